// CircleLoss_55482387529741
// MI455X (gfx1250) — compile-verified
//
#include <hip/hip_runtime.h>
#include <hip/hip_bf16.h>
#include <math.h>

typedef __attribute__((ext_vector_type(2))) float v2f;
typedef __attribute__((ext_vector_type(4))) float v4f;
typedef __attribute__((ext_vector_type(8))) float v8f;

#define BDIM   8192
#define DDIM   128
#define MARG   0.25f
#define GAMMA  256.0f
#define G2     (256.0f * 1.4426950408889634f)   // gamma * log2(e): base-2 logits
#define LN2F   0.6931471805599453f
#define NEGINF (-1.0e30f)

#define NWAVES 16                 // waves per block (512 threads)
#define TPW    (512 / NWAVES)     // column tiles per wave = 32
#define NRT    4                  // row tiles per block
#define ROWS_PER_BLOCK (NRT * 16) // 64 rows per block
#define APITCH 132                // padded LDS row stride (floats): conflict-free

// ---------------------------------------------------------------------------
// Kernel 1: row-wise L2 normalization. One wave per row, 1 float4 per lane.
// ---------------------------------------------------------------------------
__global__ __launch_bounds__(256) void norm_rows(const float* __restrict__ a,
                                                 const float* __restrict__ b,
                                                 float* __restrict__ an,
                                                 float* __restrict__ bn) {
    const float* src = (blockIdx.y == 0) ? a : b;
    float*       dst = (blockIdx.y == 0) ? an : bn;
    const int wave = threadIdx.x >> 5;
    const int lane = threadIdx.x & 31;
    const int row  = blockIdx.x * 8 + wave;

    const float4* srow = (const float4*)(src + (size_t)row * DDIM);
    float4 v = srow[lane];
    float ss = v.x * v.x + v.y * v.y + v.z * v.z + v.w * v.w;
#pragma unroll
    for (int off = 16; off >= 1; off >>= 1) ss += __shfl_xor(ss, off, 32);
    const float inv = 1.0f / fmaxf(sqrtf(ss), 1e-12f);
    float4 o; o.x = v.x * inv; o.y = v.y * inv; o.z = v.z * inv; o.w = v.w * inv;
    ((float4*)(dst + (size_t)row * DDIM))[lane] = o;
}

// ---------------------------------------------------------------------------
// Kernel 2: fused GEMM (fp32 WMMA 16x16x4) + online masked logsumexp (base-2).
// Block = 512 threads = 16 waves, owns 64 rows (4 row tiles). A is staged once
// into LDS in *fragment order* (sA[row*132 + half*64 + 2*s] = a[row][4s+2h..])
// and shared by all waves: each B tile from L2 feeds 4x32 WMMAs (L2 traffic =
// 128 blocks x 4MB = 512MB). A-frags: single-base ds_load_b128 (2 k-steps per
// load, 16-bit imm offsets), kept in-loop by an opaque-but-aligned offset
// (anti-LICM without losing the 16B alignment proof). B tiles are half-tile
// pipelined in registers. sim never hits memory.
// ---------------------------------------------------------------------------
__global__ __launch_bounds__(512) void circle_fused(const float* __restrict__ an,
                                                    const float* __restrict__ bn,
                                                    float* __restrict__ losses) {
    __shared__ float sA[ROWS_PER_BLOCK * APITCH];       // 33 KB fragment-order A
    __shared__ float smx[NWAVES][ROWS_PER_BLOCK];       // 4 KB
    __shared__ float ssm[NWAVES][ROWS_PER_BLOCK];       // 4 KB

    const int wave = threadIdx.x >> 5;
    const int lane = threadIdx.x & 31;
    const int half = lane >> 4;
    const int l16  = lane & 15;
    const int row_base = blockIdx.x * ROWS_PER_BLOCK;

    // Cooperative stage of the 64x128 A block into fragment-order LDS:
    // float4 a[row][4q..4q+3] -> (x,y) at [row][half=0][s=q], (z,w) at [half=1][s=q].
    for (int idx = threadIdx.x; idx < ROWS_PER_BLOCK * (DDIM / 4); idx += 512) {
        const int row = idx >> 5;          // 32 float4s per row
        const int q   = idx & 31;
        float4 v = *(const float4*)(an + (size_t)(row_base + row) * DDIM + 4 * q);
        *(v2f*)&sA[row * APITCH + 2 * q]      = (v2f){v.x, v.y};
        *(v2f*)&sA[row * APITCH + 64 + 2 * q] = (v2f){v.z, v.w};
    }
    __syncthreads();

    // Online base-2 logsumexp state per row tile (logits >= 0 -> mx=0 exact).
    float mx[NRT][8], sm[NRT][8];
#pragma unroll
    for (int rt = 0; rt < NRT; ++rt)
#pragma unroll
        for (int v = 0; v < 8; ++v) { mx[rt][v] = 0.f; sm[rt][v] = 0.f; }

    const int ct0  = wave * TPW;
    const int dct0 = NRT * blockIdx.x;            // diag column tile of row tile 0
    const int abase = l16 * APITCH + half * 64;   // lane's A-frag base (mult. of 4)

    // Load one K-half (16 k-steps) of a B tile: lane n holds row (ct*16+n).
    auto loadHalfB = [&](v2f (&br)[16], int ct, int kh) {
        const float* brow = bn + (size_t)(ct * 16 + l16) * DDIM + 64 * kh + 2 * half;
#pragma unroll
        for (int s = 0; s < 16; ++s) br[s] = *(const v2f*)(brow + 4 * s);
    };

    // 16 k-steps x 4 row tiles of WMMAs. A-frags: ds_load_b128 (2 k-steps each)
    // from ONE base address + constant offsets (rt*8448B + sp*16B < 64KB imm).
    auto mmaHalf = [&](const v2f (&br)[16], int sbase, v8f (&c)[NRT], int opq) {
        const v4f* ap = (const v4f*)__builtin_assume_aligned(
                            &sA[abase + 2 * sbase + opq], 16);
#pragma unroll
        for (int sp = 0; sp < 8; ++sp) {               // pairs of k-steps
#pragma unroll
            for (int rt = 0; rt < NRT; ++rt) {
                const v4f q = ap[rt * (4 * APITCH) + sp];
                const v2f a2a = {q.x, q.y};
                const v2f a2b = {q.z, q.w};
                c[rt] = __builtin_amdgcn_wmma_f32_16x16x4_f32(
                            false, a2a, false, br[2 * sp], (short)0, c[rt],
                            false, false);
                c[rt] = __builtin_amdgcn_wmma_f32_16x16x4_f32(
                            false, a2b, false, br[2 * sp + 1], (short)0, c[rt],
                            false, false);
            }
        }
    };

    // Fast path: no diagonal in this tile -- no compares.
    auto epiPlain = [&](const v8f& c, float (&mxr)[8], float (&smr)[8]) {
#pragma unroll
        for (int v = 0; v < 8; ++v) {
            const float s_ = c[v];
            const float x  = G2 * fmaxf(s_ - MARG, 0.f) * (s_ + MARG);
            const float nm = fmaxf(mxr[v], x);
            smr[v] = smr[v] * exp2f(mxr[v] - nm) + exp2f(x - nm);
            mxr[v] = nm;
        }
    };
    // Diagonal-aware path: mask the positive out of the neg-LSE (NEGINF).
    auto epiDiag = [&](const v8f& c, float (&mxr)[8], float (&smr)[8],
                       int rb, int j) {
#pragma unroll
        for (int v = 0; v < 8; ++v) {
            const float s_  = c[v];
            const bool  dia = (rb + v + 8 * half) == j;
            const float ln  = G2 * fmaxf(s_ - MARG, 0.f) * (s_ + MARG);
            const float x   = dia ? NEGINF : ln;      // exp2(NEGINF-mx) -> 0
            const float nm = fmaxf(mxr[v], x);
            smr[v] = smr[v] * exp2f(mxr[v] - nm) + exp2f(x - nm);
            mxr[v] = nm;
        }
    };

    // Half-tile software pipeline over this wave's 32 column tiles.
    v2f bl[16], bh[16];
    loadHalfB(bl, ct0, 0);
    for (int t = 0; t < TPW; ++t) {
        const int ct  = ct0 + t;
        const int ctn = (t + 1 < TPW) ? ct + 1 : ct0;

        // Opaque zero, fresh every iteration: keeps A-frag ds_loads in-loop
        // (no LICM hoist+spill). Masking low bits restores the compiler's
        // 16-byte alignment proof so the loads stay ds_load_b128.
        int opq = 0;
        asm volatile("" : "+v"(opq));
        opq &= ~3;

        loadHalfB(bh, ct, 1);                 // hi half of current tile
        v8f c[NRT];
#pragma unroll
        for (int rt = 0; rt < NRT; ++rt)
            c[rt] = (v8f){0.f, 0.f, 0.f, 0.f, 0.f, 0.f, 0.f, 0.f};
        mmaHalf(bl, 0, c, opq);               // consume lo half (k 0..63)
        loadHalfB(bl, ctn, 0);                // lo half of next tile
        mmaHalf(bh, 16, c, opq);              // consume hi half (k 64..127)

        // Epilogue: diag check is a uniform scalar branch per row tile.
        const int j = ct * 16 + l16;
#pragma unroll
        for (int rt = 0; rt < NRT; ++rt) {
            if (ct == dct0 + rt) epiDiag(c[rt], mx[rt], sm[rt],
                                         row_base + rt * 16, j);
            else                 epiPlain(c[rt], mx[rt], sm[rt]);
        }
    }

    // Combine the 16 lanes (N direction) sharing each row, per half-wave.
#pragma unroll
    for (int off = 1; off <= 8; off <<= 1) {
#pragma unroll
        for (int rt = 0; rt < NRT; ++rt) {
#pragma unroll
            for (int v = 0; v < 8; ++v) {
                const float omx = __shfl_xor(mx[rt][v], off, 32);
                const float osm = __shfl_xor(sm[rt][v], off, 32);
                const float nm  = fmaxf(mx[rt][v], omx);
                sm[rt][v] = sm[rt][v] * exp2f(mx[rt][v] - nm)
                          + osm * exp2f(omx - nm);
                mx[rt][v] = nm;
            }
        }
    }

    if (l16 == 0) {                        // lanes 0 and 16 publish 8 rows each
#pragma unroll
        for (int rt = 0; rt < NRT; ++rt)
#pragma unroll
            for (int v = 0; v < 8; ++v) {
                const int r = rt * 16 + v + 8 * half;
                smx[wave][r] = mx[rt][v];
                ssm[wave][r] = sm[rt][v];
            }
    }
    __syncthreads();

    // Combine the NWAVES wave partials, finish the loss for 64 rows.
    if (threadIdx.x < ROWS_PER_BLOCK) {
        const int r = threadIdx.x;
        const int i = row_base + r;
        float m = smx[0][r];
#pragma unroll
        for (int w = 1; w < NWAVES; ++w) m = fmaxf(m, smx[w][r]);
        float s = 0.0f;
#pragma unroll
        for (int w = 0; w < NWAVES; ++w) s += ssm[w][r] * exp2f(smx[w][r] - m);

        // s_ii = dot(a_n[i], b_n[i]) computed directly (128 FMAs).
        const float4* pa = (const float4*)(an + (size_t)i * DDIM);
        const float4* pb = (const float4*)(bn + (size_t)i * DDIM);
        float sii = 0.0f;
#pragma unroll
        for (int q = 0; q < DDIM / 4; ++q) {
            float4 x = pa[q], y = pb[q];
            sii += x.x * y.x + x.y * y.y + x.z * y.z + x.w * y.w;
        }

        const float lse_neg = (m + log2f(s)) * LN2F;              // back to base-e
        const float lp = -GAMMA * fmaxf((1.0f - MARG) - sii, 0.0f) * (sii - MARG);
        const float z  = lp + lse_neg;
        const float loss = fmaxf(z, 0.0f) + log1pf(__expf(-fabsf(z)));
        losses[i] = loss;
    }
}

// ---------------------------------------------------------------------------
// Kernel 3: deterministic fixed-order mean over 8192 row losses.
// ---------------------------------------------------------------------------
__global__ __launch_bounds__(256) void reduce_mean(const float* __restrict__ losses,
                                                   float* __restrict__ out) {
    __shared__ float red[256];
    float acc = 0.0f;
    for (int i = threadIdx.x; i < BDIM; i += 256) acc += losses[i];
    red[threadIdx.x] = acc;
    __syncthreads();
    for (int s = 128; s > 0; s >>= 1) {
        if (threadIdx.x < s) red[threadIdx.x] += red[threadIdx.x + s];
        __syncthreads();
    }
    if (threadIdx.x == 0) out[0] = red[0] * (1.0f / (float)BDIM);
}

extern "C" void kernel_launch(void* const* d_in, const int* in_sizes, int n_in,
                              void* d_out, int out_size, void* d_ws, size_t ws_size,
                              hipStream_t stream) {
    const float* ea = (const float*)d_in[0];
    const float* eb = (const float*)d_in[1];
    float* out = (float*)d_out;

    // ws layout: a_norm (4MB) | b_norm (4MB) | per-row losses (32KB)
    float* an     = (float*)d_ws;
    float* bn     = an + (size_t)BDIM * DDIM;
    float* losses = bn + (size_t)BDIM * DDIM;

    dim3 gN(BDIM / 8, 2);
    norm_rows<<<gN, 256, 0, stream>>>(ea, eb, an, bn);
    circle_fused<<<BDIM / ROWS_PER_BLOCK, 512, 0, stream>>>(an, bn, losses);
    reduce_mean<<<1, 256, 0, stream>>>(losses, out);
}